// Encoder_87282325390064
// MI455X (gfx1250) — compile-verified
//
#include <hip/hip_runtime.h>

typedef float v2f __attribute__((ext_vector_type(2)));
typedef float v8f __attribute__((ext_vector_type(8)));

#define HFEAT 64

// ---------------------------------------------------------------------------
// Zero a float region (agg + cnt are contiguous in workspace).
// ---------------------------------------------------------------------------
__global__ void zero_kernel(float* __restrict__ p, long long n) {
  long long i = (long long)blockIdx.x * blockDim.x + threadIdx.x;
  if (i < n) p[i] = 0.0f;
}

// ---------------------------------------------------------------------------
// Edge scatter: agg[dst] += feat[src] (64 floats/edge, one lane per feature),
// optionally counting in-degree. Node arrays are L2-resident (25.6MB << 192MB)
// so the f32 atomics resolve in L2.
// ---------------------------------------------------------------------------
__global__ void scatter_kernel(const float* __restrict__ feat,
                               const int*   __restrict__ edge,
                               float* __restrict__ agg,
                               float* __restrict__ cnt,
                               int E, int with_cnt) {
  long long t = (long long)blockIdx.x * blockDim.x + threadIdx.x;
  if (t >= (long long)E * HFEAT) return;
  int e = (int)(t >> 6);
  int f = (int)(t & 63);
  int src = edge[e];          // row 0 of [2,E]
  int dst = edge[E + e];      // row 1 of [2,E]
  atomicAdd(&agg[(long long)dst * HFEAT + f], feat[(long long)src * HFEAT + f]);
  if (with_cnt && f == 0) atomicAdd(&cnt[dst], 1.0f);
}

// ---------------------------------------------------------------------------
// Fused SAGE layer GEMM: out = act( (agg/max(cnt,1)) @ Wl + hin @ Wr + b )
// One wave32 computes a 16(node)x16(channel) tile with V_WMMA_F32_16X16X4_F32,
// sweeping K=64 in 16 steps per operand product (32 WMMA / tile).
//
// A (16x4 f32) per-lane layout (ISA 7.12.2): lanes 0-15 hold K=kk,kk+1 in
// v[0],v[1]; lanes 16-31 hold K=kk+2,kk+3.  B (4x16): rows striped across
// lanes within a VGPR, mirrored.  C/D: VGPR r -> row r (lanes 0-15) and
// row 8+r (lanes 16-31).
// ---------------------------------------------------------------------------
template <bool DROPOUT>
__global__ void sage_gemm_kernel(const float* __restrict__ agg,
                                 const float* __restrict__ cnt,
                                 const float* __restrict__ hin,
                                 const float* __restrict__ Wl,
                                 const float* __restrict__ Wr,
                                 const float* __restrict__ bias,
                                 const float* __restrict__ u1,
                                 float* __restrict__ hout,
                                 int N, int ntiles) {
  const int lane = threadIdx.x & 31;
  const int wave = threadIdx.x >> 5;
  const int tile = blockIdx.x * (blockDim.x >> 5) + wave;
  if (tile >= ntiles) return;            // wave-uniform: EXEC stays all-1s

  const int tm  = (tile >> 2) << 4;      // node-tile * 16 (4 channel tiles)
  const int tn  = (tile & 3) << 4;       // channel-tile * 16
  const int hi  = lane >> 4;             // half-wave: K pair select
  const int l16 = lane & 15;
  const int arow = tm + l16;             // A-matrix row (node)
  const int bcol = tn + l16;             // B-matrix col (channel)
  const float invc = 1.0f / fmaxf(cnt[arow], 1.0f);

  const float* __restrict__ aptr = agg + (long long)arow * HFEAT + 2 * hi;
  const float* __restrict__ hptr = hin + (long long)arow * HFEAT + 2 * hi;
  const float* __restrict__ wl   = Wl + 2 * hi * HFEAT + bcol;
  const float* __restrict__ wr   = Wr + 2 * hi * HFEAT + bcol;

  v8f acc = {0.f, 0.f, 0.f, 0.f, 0.f, 0.f, 0.f, 0.f};

#pragma unroll
  for (int kk = 0; kk < HFEAT; kk += 4) {   // mean @ Wl
    v2f a, b;
    a.x = aptr[kk] * invc;
    a.y = aptr[kk + 1] * invc;
    b.x = wl[kk * HFEAT];
    b.y = wl[(kk + 1) * HFEAT];
    acc = __builtin_amdgcn_wmma_f32_16x16x4_f32(false, a, false, b,
                                                (short)0, acc, false, false);
  }
#pragma unroll
  for (int kk = 0; kk < HFEAT; kk += 4) {   // hin @ Wr (root weight)
    v2f a, b;
    a.x = hptr[kk];
    a.y = hptr[kk + 1];
    b.x = wr[kk * HFEAT];
    b.y = wr[(kk + 1) * HFEAT];
    acc = __builtin_amdgcn_wmma_f32_16x16x4_f32(false, a, false, b,
                                                (short)0, acc, false, false);
  }

  const float bv = bias[bcol];
#pragma unroll
  for (int r = 0; r < 8; ++r) {
    const int orow = tm + r + 8 * hi;       // D layout: row r / row 8+r
    float z = fmaxf(acc[r] + bv, 0.0f);     // ReLU
    if (DROPOUT) {
      const float u = u1[(long long)orow * HFEAT + bcol];
      z = (u > 0.5f) ? (z * 2.0f) : 0.0f;   // mask/(1-P), P=0.5
    }
    hout[(long long)orow * HFEAT + bcol] = z;
  }
}

// ---------------------------------------------------------------------------
extern "C" void kernel_launch(void* const* d_in, const int* in_sizes, int n_in,
                              void* d_out, int out_size, void* d_ws, size_t ws_size,
                              hipStream_t stream) {
  const float* x   = (const float*)d_in[0];
  const int*   edg = (const int*)d_in[1];
  const float* u1  = (const float*)d_in[2];
  const float* W1l = (const float*)d_in[3];
  const float* W1r = (const float*)d_in[4];
  const float* b1  = (const float*)d_in[5];
  const float* W2l = (const float*)d_in[6];
  const float* W2r = (const float*)d_in[7];
  const float* b2  = (const float*)d_in[8];
  float* out = (float*)d_out;

  const int N = in_sizes[0] / HFEAT;   // 100000
  const int E = in_sizes[1] / 2;       // 1200000

  float* agg = (float*)d_ws;                 // [N,64]
  float* cnt = agg + (size_t)N * HFEAT;      // [N]
  float* h1  = cnt + N;                      // [N,64]

  const long long nzero = (long long)N * HFEAT + N;   // agg + cnt contiguous
  const long long nscat = (long long)E * HFEAT;
  const int ntiles = ((N + 15) / 16) * (HFEAT / 16);
  const int gblocks = (ntiles + 7) / 8;               // 8 waves / 256-thread block

  // Layer 1
  zero_kernel<<<(int)((nzero + 255) / 256), 256, 0, stream>>>(agg, nzero);
  scatter_kernel<<<(int)((nscat + 255) / 256), 256, 0, stream>>>(x, edg, agg, cnt, E, 1);
  sage_gemm_kernel<true><<<gblocks, 256, 0, stream>>>(agg, cnt, x, W1l, W1r, b1,
                                                      u1, h1, N, ntiles);
  // Layer 2 (degree counts unchanged; reuse cnt)
  zero_kernel<<<(int)(((long long)N * HFEAT + 255) / 256), 256, 0, stream>>>(
      agg, (long long)N * HFEAT);
  scatter_kernel<<<(int)((nscat + 255) / 256), 256, 0, stream>>>(h1, edg, agg, cnt, E, 0);
  sage_gemm_kernel<false><<<gblocks, 256, 0, stream>>>(agg, cnt, h1, W2l, W2r, b2,
                                                       nullptr, out, N, ntiles);
}